// GINEConvMultiEdgeset_40853728920023
// MI455X (gfx1250) — compile-verified
//
#include <hip/hip_runtime.h>
#include <hip/hip_bf16.h>

typedef float v2f __attribute__((ext_vector_type(2)));
typedef float v8f __attribute__((ext_vector_type(8)));

#define NNODES 100000
#define NEDGES 400000
#define DIMD   64
#define NCH    4          // R*C = 1*4
#define NROWS  (NCH * NNODES)   // 400000 rows through the MLP

__device__ __forceinline__ float gelu_exact(float v) {
    // torch F.gelu default: 0.5*x*(1+erf(x/sqrt(2)))
    return 0.5f * v * (1.0f + erff(v * 0.70710678118654752f));
}

// ---------------- Kernel 0: agg = (1+eps) * x  (float4 vectorized) -------------
__global__ void init_agg(const float* __restrict__ x,
                         const float* __restrict__ eps,
                         float* __restrict__ agg, int total4) {
    int i = blockIdx.x * blockDim.x + threadIdx.x;
    if (i >= total4) return;
    float s = 1.0f + eps[0];
    float4 v = ((const float4*)x)[i];
    v.x *= s; v.y *= s; v.z *= s; v.w *= s;
    ((float4*)agg)[i] = v;
}

// ---------------- Kernel 1: edge gather + gelu + scatter-add -------------------
// one wave32 per edge; lane owns float2 of the 64-wide feature
__global__ void edge_scatter(const float* __restrict__ x,
                             const int*   __restrict__ eidx,   // [2,E]
                             const int*   __restrict__ eattr,  // [E,3]
                             const float* __restrict__ ew,     // [C,E]
                             const float* __restrict__ be0,
                             const float* __restrict__ be1,
                             const float* __restrict__ be2,
                             float* __restrict__ agg) {
    int e = (blockIdx.x * blockDim.x + threadIdx.x) >> 5;
    if (e >= NEDGES) return;
    int lane = threadIdx.x & 31;
    int d = lane << 1;

    int src = eidx[e];
    int dst = eidx[NEDGES + e];
    int a0 = eattr[e * 3 + 0];
    int a1 = eattr[e * 3 + 1];
    int a2 = eattr[e * 3 + 2];

    float ex = be0[a0 * DIMD + d]     + be1[a1 * DIMD + d]     + be2[a2 * DIMD + d];
    float ey = be0[a0 * DIMD + d + 1] + be1[a1 * DIMD + d + 1] + be2[a2 * DIMD + d + 1];

#pragma unroll
    for (int c = 0; c < NCH; ++c) {
        float w = ew[c * NEDGES + e];
        const float2 xv = *(const float2*)(x + (size_t)(c * NNODES + src) * DIMD + d);
        float m0 = gelu_exact(xv.x + ex) * w;
        float m1 = gelu_exact(xv.y + ey) * w;
        float* p = agg + (size_t)(c * NNODES + dst) * DIMD + d;
        atomicAdd(p,     m0);   // global_atomic_add_f32, L2-resident target
        atomicAdd(p + 1, m1);
    }
}

// ---------------- Kernel 2: fused MLP via V_WMMA_F32_16X16X4_F32 ---------------
// out = gelu(agg @ W1 + b1) @ W2 + b2 ; one wave per 16-row tile, 4 waves/block
__global__ __launch_bounds__(128)
void mlp_wmma(const float* __restrict__ agg,
              const float* __restrict__ W1, const float* __restrict__ b1,
              const float* __restrict__ W2, const float* __restrict__ b2,
              float* __restrict__ out) {
    // per-wave 16x128 gelu(h1) staging; stride 132 -> 4-bank row skew
    __shared__ float h1s[4][16][132];

    int wave = threadIdx.x >> 5;
    int lane = threadIdx.x & 31;
    int half = lane >> 4;      // 0: K even pair, 1: K odd pair (A/B frag layout)
    int mr   = lane & 15;
    size_t row0 = ((size_t)blockIdx.x * 4 + wave) * 16;

    // Preload all A fragments for the 16x64 tile.
    // ISA 16x4 f32 A layout: lane<16 -> A[M=lane][k],A[M=lane][k+1];
    //                        lane>=16 -> A[M=lane-16][k+2],[k+3]
    v2f afrag[16];
    const float* arow = agg + (row0 + mr) * DIMD;
#pragma unroll
    for (int kk = 0; kk < 16; ++kk) {
        int col = kk * 4 + half * 2;
        afrag[kk].x = arow[col];
        afrag[kk].y = arow[col + 1];
    }

    // ---- GEMM1: (16x64) @ W1(64x128) + b1, then exact GELU -> LDS ----
#pragma unroll
    for (int nt = 0; nt < 8; ++nt) {
        float bias = b1[nt * 16 + mr];            // C layout: lane&15 == N
        v8f acc = {bias, bias, bias, bias, bias, bias, bias, bias};
#pragma unroll
        for (int kk = 0; kk < 16; ++kk) {
            int k = kk * 4 + half * 2;
            v2f bf;                                // B frag: W1[k][n], W1[k+1][n]
            bf.x = W1[k * 128 + nt * 16 + mr];
            bf.y = W1[(k + 1) * 128 + nt * 16 + mr];
            acc = __builtin_amdgcn_wmma_f32_16x16x4_f32(
                false, afrag[kk], false, bf, (short)0, acc, false, false);
        }
#pragma unroll
        for (int vg = 0; vg < 8; ++vg) {
            int m = vg + half * 8;                 // C layout: M = vgpr + 8*(lane>=16)
            h1s[wave][m][nt * 16 + mr] = gelu_exact(acc[vg]);
        }
    }
    __syncthreads();   // C-layout -> A-layout re-shuffle goes through LDS

    // ---- GEMM2: (16x128) @ W2(128x64) + b2 -> out ----
#pragma unroll
    for (int nt = 0; nt < 4; ++nt) {
        float bias = b2[nt * 16 + mr];
        v8f acc = {bias, bias, bias, bias, bias, bias, bias, bias};
#pragma unroll
        for (int kk = 0; kk < 32; ++kk) {
            int k = kk * 4 + half * 2;
            v2f af;                                // A frag from staged gelu(h1)
            af.x = h1s[wave][mr][k];
            af.y = h1s[wave][mr][k + 1];
            v2f bf;
            bf.x = W2[k * 64 + nt * 16 + mr];
            bf.y = W2[(k + 1) * 64 + nt * 16 + mr];
            acc = __builtin_amdgcn_wmma_f32_16x16x4_f32(
                false, af, false, bf, (short)0, acc, false, false);
        }
#pragma unroll
        for (int vg = 0; vg < 8; ++vg) {
            int m = vg + half * 8;
            out[(row0 + m) * DIMD + nt * 16 + mr] = acc[vg];
        }
    }
}

extern "C" void kernel_launch(void* const* d_in, const int* in_sizes, int n_in,
                              void* d_out, int out_size, void* d_ws, size_t ws_size,
                              hipStream_t stream) {
    const float* x     = (const float*)d_in[0];   // [1,4,N,64]
    const int*   eidx  = (const int*)  d_in[1];   // [2,E]
    const int*   eattr = (const int*)  d_in[2];   // [E,3]
    const float* ew    = (const float*)d_in[3];   // [1,4,E,1]
    const float* be0   = (const float*)d_in[4];
    const float* be1   = (const float*)d_in[5];
    const float* be2   = (const float*)d_in[6];
    const float* W1    = (const float*)d_in[7];   // [64,128]
    const float* b1    = (const float*)d_in[8];
    const float* W2    = (const float*)d_in[9];   // [128,64]
    const float* b2    = (const float*)d_in[10];
    const float* eps   = (const float*)d_in[11];

    float* agg = (float*)d_ws;                    // [4,N,64] fp32 scratch (~102 MB, L2-resident)
    float* outp = (float*)d_out;

    // 1) agg = (1+eps)*x
    int total4 = (NCH * NNODES * DIMD) / 4;       // 6.4M float4
    init_agg<<<(total4 + 255) / 256, 256, 0, stream>>>(x, eps, agg, total4);

    // 2) edge message + scatter-add (one wave per edge, 8 edges per 256-thread block)
    edge_scatter<<<NEDGES / 8, 256, 0, stream>>>(x, eidx, eattr, ew, be0, be1, be2, agg);

    // 3) fused WMMA MLP: 400000 rows / 16 per wave / 4 waves per block
    mlp_wmma<<<NROWS / 16 / 4, 128, 0, stream>>>(agg, W1, b1, W2, b2, outp);
}